// OCP_35476429865433
// MI455X (gfx1250) — compile-verified
//
#include <hip/hip_runtime.h>
#include <math.h>

// ---------------------------------------------------------------------------
// Batched SQP/Riccati OCP evaluation for MI455X (gfx1250, wave32, WMMA).
// One workgroup per batch element; all heavy GEMMs via V_WMMA_F32_16X16X4_F32.
// ---------------------------------------------------------------------------

#define DEVINL __device__ __forceinline__

constexpr int NX  = 128;
constexpr int NU  = 64;
constexpr int NH  = 32;
constexpr int NB  = 128;
constexpr int NXU = NX + NU;   // 192
constexpr float GAMMA_DISC = 0.99f;
constexpr int N_ITER = 2;
constexpr int BLOCK  = 256;    // 8 wave32 waves
constexpr int NWAVES = BLOCK / 32;

typedef __attribute__((ext_vector_type(2))) float v2f;
typedef __attribute__((ext_vector_type(8))) float v8f;

// ---------------- per-batch global workspace layout (floats) ---------------
constexpr int WS_XB    = 0;                          // (NH+1)*NX = 4224
constexpr int WS_UB    = WS_XB    + (NH+1)*NX;       // NH*NU     = 2048
constexpr int WS_LB    = WS_UB    + NH*NU;           // (NH+1)*NX = 4224
constexpr int WS_GM    = WS_LB    + (NH+1)*NX;       // NU
constexpr int WS_LXU   = WS_GM    + NU;              // NH*NXU    = 6144
constexpr int WS_LXN   = WS_LXU   + NH*NXU;          // NX
constexpr int WS_XRES  = WS_LXN   + NX;              // NH*NX     = 4096
constexpr int WS_X0RES = WS_XRES  + NH*NX;           // NX
constexpr int WS_LS    = WS_X0RES + NX;              // NH
constexpr int WS_LN    = WS_LS    + NH;              // 1 (pad to 32)
constexpr int WS_PS    = WS_LN    + 32;              // NH*NX*NX  = 524288
constexpr int WS_PP    = WS_PS    + NH*NX*NX;        // NH*NX     = 4096
constexpr int WS_KS    = WS_PP    + NH*NX;           // NH*NU*NX  = 262144
constexpr int WS_KK    = WS_KS    + NH*NU*NX;        // NH*NU     = 2048
constexpr int WS_PER_BATCH = WS_KK + NH*NU;          // 813696 floats / batch

// ---------------- dynamic LDS layout (floats) ------------------------------
constexpr int L_P   = 0;                 // 128x128 value Hessian carry
constexpr int L_T   = L_P   + NX*NX;     // 128x192 PF  (also Dxs / solve RHS)
constexpr int L_HXU = L_T   + NX*NXU;    // 128x64
constexpr int L_HUU = L_HXU + NX*NU;     // 64x65 (padded)
constexpr int L_VEC = L_HUU + NU*65;
constexpr int L_W    = L_VEC;            // 128
constexpr int L_HX   = L_VEC + 128;      // 128
constexpr int L_HU   = L_VEC + 256;      // 64
constexpr int L_PV   = L_VEC + 320;      // 128
constexpr int L_DX   = L_VEC + 448;      // 128
constexpr int L_DU   = L_VEC + 576;      // 64
constexpr int L_DXN  = L_VEC + 640;      // 128
constexpr int L_DX0  = L_VEC + 768;      // 128
constexpr int L_DU0  = L_VEC + 896;      // 64
constexpr int L_DGM  = L_VEC + 960;      // 64
constexpr int L_DL0  = L_VEC + 1024;     // 128
constexpr int L_QV   = L_VEC + 1152;     // 128
constexpr int L_DT   = L_VEC + 1280;     // 128
constexpr int L_RED  = L_VEC + 1408;     // 256
constexpr int SMEM_FLOATS = L_VEC + 1664;   // 54976 floats = 219904 bytes

// ---------------------------------------------------------------------------
DEVINL float fld(const float* M, int ld, int r, int c, int tr) {
  return tr ? M[c * ld + r] : M[r * ld + c];
}

// C[M x N] (row-major, ldc) = alpha * op(A)(MxK) * op(B)(KxN) + beta * C
// M, N multiples of 16; K multiple of 4. Uses V_WMMA_F32_16X16X4_F32.
__device__ void gemm(float* C, int ldc, int M, int N, int K,
                     const float* A, int lda, int tA,
                     const float* B, int ldb, int tB,
                     float alpha, float beta)
{
  const int lane = threadIdx.x & 31;
  const int wave = threadIdx.x >> 5;
  const int tN = N >> 4;
  const int nTiles = (M >> 4) * tN;
  const int mf = lane & 15;            // A row / B,C col within tile
  const int kh = (lane >> 4) << 1;     // K sub-offset: 0 or 2
  for (int t = wave; t < nTiles; t += NWAVES) {
    const int r0 = (t / tN) << 4;
    const int c0 = (t % tN) << 4;
    v8f acc = {0.f, 0.f, 0.f, 0.f, 0.f, 0.f, 0.f, 0.f};
    for (int k = 0; k < K; k += 4) {
      v2f a, b;
      a.x = fld(A, lda, r0 + mf, k + kh + 0, tA);
      a.y = fld(A, lda, r0 + mf, k + kh + 1, tA);
      b.x = fld(B, ldb, k + kh + 0, c0 + mf, tB);
      b.y = fld(B, ldb, k + kh + 1, c0 + mf, tB);
      acc = __builtin_amdgcn_wmma_f32_16x16x4_f32(false, a, false, b,
                                                  (short)0, acc, false, false);
    }
    const int mb = r0 + ((lane >> 4) << 3);
    const int nn = c0 + mf;
#pragma unroll
    for (int r = 0; r < 8; ++r) {
      float* cp = C + (size_t)(mb + r) * ldc + nn;
      float oldv = (beta != 0.f) ? beta * (*cp) : 0.f;
      *cp = oldv + alpha * acc[r];
    }
  }
}

// y[i] = (acc ? y[i] : 0) + alpha * sum_j op(M)[i,j] x[j]
__device__ void matvec(float* y, const float* M, int ld, int tr,
                       const float* x, int Mrows, int Kcols,
                       float alpha, int accumulate)
{
  for (int i = threadIdx.x; i < Mrows; i += BLOCK) {
    float s = 0.f;
    for (int j = 0; j < Kcols; ++j) s += fld(M, ld, i, j, tr) * x[j];
    y[i] = (accumulate ? y[i] : 0.f) + alpha * s;
  }
}

__device__ void vcopy(float* dst, const float* src, int n) {
  for (int i = threadIdx.x; i < n; i += BLOCK) dst[i] = src[i];
}

__device__ float block_reduce(float v, float* red) {
  const int tid = threadIdx.x;
  red[tid] = v;
  __syncthreads();
  for (int s = BLOCK / 2; s > 0; s >>= 1) {
    if (tid < s) red[tid] += red[tid + s];
    __syncthreads();
  }
  float r = red[0];
  __syncthreads();
  return r;
}

// ---------------------------------------------------------------------------
// KKT evaluation for one batch element.
// ---------------------------------------------------------------------------
__device__ void eval_kkt(float* sm,
                         const float* Ag, const float* Bg, const float* Qxg,
                         const float* Rug, const float* Qfg, const float* xgg,
                         const float* x0vec,
                         const float* xb, const float* ub, const float* lb,
                         const float* gmb, int add_gmm,
                         float* lxu, float* lxN, float* xres, float* x0res,
                         float* ls, float* lNp)
{
  float* Dxs = sm + L_T;    // NH x NX
  float* dT  = sm + L_DT;
  float* qv  = sm + L_QV;

  for (int idx = threadIdx.x; idx < NH * NX; idx += BLOCK)
    Dxs[idx] = xb[idx] - xgg[idx & (NX - 1)];
  for (int i = threadIdx.x; i < NX; i += BLOCK)
    dT[i] = xb[NH * NX + i] - xgg[i];
  __syncthreads();

  // lx = Dxs*Qx ; lu = U*Ru   (pre-lambda, needed for stage costs)
  gemm(lxu,      NXU, NH, NX, NX, Dxs, NX, 0, Qxg, NX, 0, 1.f, 0.f);
  gemm(lxu + NX, NXU, NH, NU, NU, ub,  NU, 0, Rug, NU, 0, 1.f, 0.f);
  // qv = dT . Qf
  matvec(qv, Qfg, NX, 1, dT, NX, NX, 1.f, 0);
  __syncthreads();

  // stage costs ls, terminal lN, lxN
  for (int n = threadIdx.x; n < NH; n += BLOCK) {
    float s = 0.f;
    for (int i = 0; i < NX; ++i) s += Dxs[n * NX + i] * lxu[n * NXU + i];
    for (int j = 0; j < NU; ++j) s += ub[n * NU + j] * lxu[n * NXU + NX + j];
    ls[n] = 0.5f * s;
  }
  {
    float part = 0.f;
    for (int i = threadIdx.x; i < NX; i += BLOCK) part += qv[i] * dT[i];
    float tot = block_reduce(part, sm + L_RED);
    if (threadIdx.x == 0) *lNp = 0.5f * tot;
  }
  for (int i = threadIdx.x; i < NX; i += BLOCK)
    lxN[i] = qv[i] - lb[NH * NX + i];
  __syncthreads();

  // lxu += lmd[1:] * [A | B]
  gemm(lxu,      NXU, NH, NX, NX, lb + NX, NX, 0, Ag, NX, 0, 1.f, 1.f);
  __syncthreads();
  gemm(lxu + NX, NXU, NH, NU, NX, lb + NX, NX, 0, Bg, NU, 0, 1.f, 1.f);
  __syncthreads();
  for (int idx = threadIdx.x; idx < NH * NX; idx += BLOCK) {
    int n = idx >> 7, i = idx & (NX - 1);
    lxu[n * NXU + i] -= lb[n * NX + i];
  }
  if (add_gmm)
    for (int j = threadIdx.x; j < NU; j += BLOCK) lxu[NX + j] += gmb[j];
  for (int i = threadIdx.x; i < NX; i += BLOCK) x0res[i] = xb[i] - x0vec[i];
  __syncthreads();

  // xres = X*A^T + U*B^T - x[1:]
  gemm(xres, NX, NH, NX, NX, xb, NX, 0, Ag, NX, 1, 1.f, 0.f);
  __syncthreads();
  gemm(xres, NX, NH, NX, NU, ub, NU, 0, Bg, NU, 1, 1.f, 1.f);
  __syncthreads();
  for (int idx = threadIdx.x; idx < NH * NX; idx += BLOCK)
    xres[idx] -= xb[NX + idx];
  __syncthreads();
}

// ---------------------------------------------------------------------------
// Riccati backward recursion (32 stages). qmode: stash dgmm/dlmd0 at stage 0.
// ---------------------------------------------------------------------------
__device__ void riccati_bwd(float* sm,
                            const float* Ag, const float* Bg, const float* Qxg,
                            const float* Rug, const float* Qfg,
                            const float* lxu, const float* lxN, const float* xres,
                            float* Psg, float* psg, float* Ksg, float* ksg,
                            int qmode)
{
  float* P   = sm + L_P;
  float* T   = sm + L_T;
  float* Hxu = sm + L_HXU;
  float* Huu = sm + L_HUU;   // ld 65
  float* w   = sm + L_W;
  float* hx  = sm + L_HX;
  float* hu  = sm + L_HU;
  float* pv  = sm + L_PV;

  vcopy(P, Qfg, NX * NX);
  vcopy(pv, lxN, NX);
  __syncthreads();

  for (int n = NH - 1; n >= 0; --n) {
    const float* lxn = lxu + n * NXU;
    const float* rn  = xres + n * NX;

    // T = P*[A|B]
    gemm(T,      NXU, NX, NX, NX, P, NX, 0, Ag, NX, 0, 1.f, 0.f);
    gemm(T + NX, NXU, NX, NU, NX, P, NX, 0, Bg, NU, 0, 1.f, 0.f);
    vcopy(w, pv, NX);
    __syncthreads();
    // w = pv + P*xres[n]
    matvec(w, P, NX, 0, rn, NX, NX, 1.f, 1);
    __syncthreads();

    // Huu = Ru + B^T*(PB) ; Hxu = A^T*(PB)
    for (int idx = threadIdx.x; idx < NU * NU; idx += BLOCK)
      Huu[(idx / NU) * 65 + (idx % NU)] = Rug[idx];
    vcopy(hx, lxn, NX);
    vcopy(hu, lxn + NX, NU);
    __syncthreads();
    gemm(Huu, 65, NU, NU, NX, Bg, NU, 1, T + NX, NXU, 0, 1.f, 1.f);
    gemm(Hxu, NU, NX, NU, NX, Ag, NX, 1, T + NX, NXU, 0, 1.f, 0.f);
    matvec(hx, Ag, NX, 1, w, NX, NX, 1.f, 1);
    matvec(hu, Bg, NU, 1, w, NU, NX, 1.f, 1);
    __syncthreads();

    // P <- Hxx = Qx + A^T*(PA)   (P no longer read beyond this point)
    vcopy(P, Qxg, NX * NX);
    __syncthreads();
    gemm(P, NX, NX, NX, NX, Ag, NX, 1, T, NXU, 0, 1.f, 1.f);
    __syncthreads();

    if (qmode && n == 0) {
      // dgmm = -(hu + Huu du0 + Hxu^T dx0); dlmd0 = hx + Hxx dx0 + Hxu du0
      float* dx0 = sm + L_DX0;
      float* du0 = sm + L_DU0;
      for (int j = threadIdx.x; j < NU; j += BLOCK) {
        float s = hu[j];
        for (int t = 0; t < NU; ++t) s += Huu[j * 65 + t] * du0[t];
        for (int i = 0; i < NX; ++i) s += Hxu[i * NU + j] * dx0[i];
        sm[L_DGM + j] = -s;
      }
      for (int i = threadIdx.x; i < NX; i += BLOCK) {
        float s = hx[i];
        for (int j = 0; j < NX; ++j) s += P[i * NX + j] * dx0[j];
        for (int j = 0; j < NU; ++j) s += Hxu[i * NU + j] * du0[j];
        sm[L_DL0 + i] = s;
      }
      __syncthreads();
    }

    // Cholesky of Huu (SPD), in place
    for (int j = 0; j < NU; ++j) {
      if (threadIdx.x == 0) {
        float d = Huu[j * 65 + j];
        for (int t = 0; t < j; ++t) d -= Huu[j * 65 + t] * Huu[j * 65 + t];
        Huu[j * 65 + j] = sqrtf(fmaxf(d, 1e-30f));
      }
      __syncthreads();
      const float dj = Huu[j * 65 + j];
      for (int i = j + 1 + threadIdx.x; i < NU; i += BLOCK) {
        float s = Huu[i * 65 + j];
        for (int t = 0; t < j; ++t) s -= Huu[i * 65 + t] * Huu[j * 65 + t];
        Huu[i * 65 + j] = s / dj;
      }
      __syncthreads();
    }

    // RHS R = [Hux | hu]  (64 x 129, ld 132) reusing T
    float* R = T;
    constexpr int NR = NX + 1, LR = 132;
    for (int idx = threadIdx.x; idx < NU * NR; idx += BLOCK) {
      int j = idx / NR, c = idx % NR;
      R[j * LR + c] = (c < NX) ? Hxu[c * NU + j] : hu[j];
    }
    __syncthreads();
    // forward solve L y = R
    for (int j = 0; j < NU; ++j) {
      const float inv = 1.f / Huu[j * 65 + j];
      for (int c = threadIdx.x; c < NR; c += BLOCK) R[j * LR + c] *= inv;
      __syncthreads();
      for (int idx = threadIdx.x; idx < (NU - 1 - j) * NR; idx += BLOCK) {
        int i = j + 1 + idx / NR, c = idx % NR;
        R[i * LR + c] -= Huu[i * 65 + j] * R[j * LR + c];
      }
      __syncthreads();
    }
    // backward solve L^T s = y
    for (int j = NU - 1; j >= 0; --j) {
      const float inv = 1.f / Huu[j * 65 + j];
      for (int c = threadIdx.x; c < NR; c += BLOCK) R[j * LR + c] *= inv;
      __syncthreads();
      for (int idx = threadIdx.x; idx < j * NR; idx += BLOCK) {
        int i = idx / NR, c = idx % NR;
        R[i * LR + c] -= Huu[j * 65 + i] * R[j * LR + c];
      }
      __syncthreads();
    }

    // K = -S_K, k = -s_k
    float* Kg = Ksg + (size_t)n * NU * NX;
    for (int idx = threadIdx.x; idx < NU * NX; idx += BLOCK)
      Kg[idx] = -R[(idx / NX) * LR + (idx % NX)];
    for (int j = threadIdx.x; j < NU; j += BLOCK)
      ksg[n * NU + j] = -R[j * LR + NX];

    // Pn = Hxx - Hxu * S_K
    gemm(P, NX, NX, NX, NU, Hxu, NU, 0, R, LR, 0, -1.f, 1.f);
    __syncthreads();
    // symmetrize
    for (int idx = threadIdx.x; idx < NX * NX; idx += BLOCK) {
      int i = idx >> 7, jj = idx & (NX - 1);
      if (i < jj) {
        float a2 = 0.5f * (P[i * NX + jj] + P[jj * NX + i]);
        P[i * NX + jj] = a2;
        P[jj * NX + i] = a2;
      }
    }
    // pn = hx - Hxu * s_k
    for (int i = threadIdx.x; i < NX; i += BLOCK) {
      float s = hx[i];
      for (int j = 0; j < NU; ++j) s -= Hxu[i * NU + j] * R[j * LR + NX];
      pv[i] = s;
    }
    __syncthreads();
    vcopy(Psg + (size_t)n * NX * NX, P, NX * NX);
    vcopy(psg + n * NX, pv, NX);
    __syncthreads();
  }
}

// ---------------------------------------------------------------------------
// Forward rollout from stage n0 (dx preloaded at sm[L_DX]); applies updates.
// ---------------------------------------------------------------------------
__device__ void rollout(float* sm,
                        const float* Ag, const float* Bg, const float* Qfg,
                        float* xb, float* ub, float* lb,
                        const float* xres, const float* lxN,
                        const float* Psg, const float* psg,
                        const float* Ksg, const float* ksg, int n0)
{
  float* dx  = sm + L_DX;
  float* du  = sm + L_DU;
  float* dxn = sm + L_DXN;
  for (int n = n0; n < NH; ++n) {
    const float* Kg = Ksg + (size_t)n * NU * NX;
    const float* Pg = Psg + (size_t)n * NX * NX;
    if (n + 1 < NH) {   // prefetch next stage's gain/Hessian
      __builtin_prefetch(Ksg + (size_t)(n + 1) * NU * NX, 0, 1);
      __builtin_prefetch(Psg + (size_t)(n + 1) * NX * NX, 0, 1);
    }
    for (int j = threadIdx.x; j < NU; j += BLOCK) {
      float s = ksg[n * NU + j];
      for (int c = 0; c < NX; ++c) s += Kg[j * NX + c] * dx[c];
      du[j] = s;
    }
    __syncthreads();
    for (int i = threadIdx.x; i < NX; i += BLOCK) {
      float dl = psg[n * NX + i];
      for (int c = 0; c < NX; ++c) dl += Pg[i * NX + c] * dx[c];
      lb[n * NX + i] += dl;
      xb[n * NX + i] += dx[i];
      float s = xres[n * NX + i];
      for (int c = 0; c < NX; ++c) s += Ag[i * NX + c] * dx[c];
      for (int c = 0; c < NU; ++c) s += Bg[i * NU + c] * du[c];
      dxn[i] = s;
    }
    for (int j = threadIdx.x; j < NU; j += BLOCK) ub[n * NU + j] += du[j];
    __syncthreads();
    vcopy(dx, dxn, NX);
    __syncthreads();
  }
  for (int i = threadIdx.x; i < NX; i += BLOCK) {
    float s = lxN[i];
    for (int c = 0; c < NX; ++c) s += Qfg[i * NX + c] * dx[c];
    lb[NH * NX + i] += s;
    xb[NH * NX + i] += dx[i];
  }
  __syncthreads();
}

__device__ float lagrangian(float* sm, const float* ls, const float* lNp,
                            const float* lb, const float* x0res,
                            const float* xres, float extra)
{
  float part = 0.f;
  for (int n = threadIdx.x; n < NH; n += BLOCK) part += ls[n];
  for (int i = threadIdx.x; i < NX; i += BLOCK) part += lb[i] * x0res[i];
  for (int idx = threadIdx.x; idx < NH * NX; idx += BLOCK)
    part += lb[NX + idx] * xres[idx];
  float tot = block_reduce(part, sm + L_RED);
  return tot + *lNp + extra;
}

__device__ float stage_cost0(float* sm, const float* Qxg, const float* Rug,
                             const float* xgg, const float* x0b, const float* u0b)
{
  float* d = sm + L_DT;
  for (int i = threadIdx.x; i < NX; i += BLOCK) d[i] = x0b[i] - xgg[i];
  __syncthreads();
  float part = 0.f;
  for (int i = threadIdx.x; i < NX; i += BLOCK) {
    float s = 0.f;
    for (int j = 0; j < NX; ++j) s += Qxg[i * NX + j] * d[j];
    part += d[i] * s;
  }
  for (int i = threadIdx.x; i < NU; i += BLOCK) {
    float s = 0.f;
    for (int j = 0; j < NU; ++j) s += Rug[i * NU + j] * u0b[j];
    part += u0b[i] * s;
  }
  return 0.5f * block_reduce(part, sm + L_RED);
}

// ---------------------------------------------------------------------------
__global__ __launch_bounds__(BLOCK)
void ocp_sqp_kernel(const float* __restrict__ x0, const float* __restrict__ x1,
                    const float* __restrict__ u0, const float* __restrict__ x_in,
                    const float* __restrict__ u_in, const float* __restrict__ lmd_in,
                    const float* __restrict__ gmm_in,
                    const float* __restrict__ Ag, const float* __restrict__ Bg,
                    const float* __restrict__ Qxg, const float* __restrict__ Rug,
                    const float* __restrict__ Qfg, const float* __restrict__ xgg,
                    float* __restrict__ out, float* __restrict__ ws)
{
  extern __shared__ float sm[];
  const int b = blockIdx.x;
  float* wsb  = ws + (size_t)b * WS_PER_BATCH;
  float* xb    = wsb + WS_XB;
  float* ub    = wsb + WS_UB;
  float* lb    = wsb + WS_LB;
  float* gmb   = wsb + WS_GM;
  float* lxu   = wsb + WS_LXU;
  float* lxN   = wsb + WS_LXN;
  float* xres  = wsb + WS_XRES;
  float* x0res = wsb + WS_X0RES;
  float* ls    = wsb + WS_LS;
  float* lNp   = wsb + WS_LN;
  float* Psg   = wsb + WS_PS;
  float* psg   = wsb + WS_PP;
  float* Ksg   = wsb + WS_KS;
  float* ksg   = wsb + WS_KK;
  const float* x0b = x0 + b * NX;
  const float* x1b = x1 + b * NX;
  const float* u0b = u0 + b * NU;

  // gather (stage, batch, dim) -> per-batch contiguous
  for (int idx = threadIdx.x; idx < (NH + 1) * NX; idx += BLOCK) {
    int n = idx / NX, i = idx % NX;
    xb[idx] = x_in[((size_t)n * NB + b) * NX + i];
    lb[idx] = lmd_in[((size_t)n * NB + b) * NX + i];
  }
  for (int idx = threadIdx.x; idx < NH * NU; idx += BLOCK) {
    int n = idx / NU, j = idx % NU;
    ub[idx] = u_in[((size_t)n * NB + b) * NU + j];
  }
  for (int j = threadIdx.x; j < NU; j += BLOCK) gmb[j] = gmm_in[b * NU + j];
  __syncthreads();

  const float L0 = stage_cost0(sm, Qxg, Rug, xgg, x0b, u0b);

  // ---------------- Q-iterations about x0 ----------------
  for (int it = 0; it < N_ITER; ++it) {
    eval_kkt(sm, Ag, Bg, Qxg, Rug, Qfg, xgg, x0b, xb, ub, lb, gmb, 1,
             lxu, lxN, xres, x0res, ls, lNp);
    for (int i = threadIdx.x; i < NX; i += BLOCK) sm[L_DX0 + i] = -x0res[i];
    for (int j = threadIdx.x; j < NU; j += BLOCK) sm[L_DU0 + j] = -(ub[j] - u0b[j]);
    __syncthreads();
    riccati_bwd(sm, Ag, Bg, Qxg, Rug, Qfg, lxu, lxN, xres,
                Psg, psg, Ksg, ksg, 1);
    // stage-0 special update + dx1 = A dx0 + B du0 + xres[0]
    for (int i = threadIdx.x; i < NX; i += BLOCK) {
      float s = xres[i];
      for (int c = 0; c < NX; ++c) s += Ag[i * NX + c] * sm[L_DX0 + c];
      for (int c = 0; c < NU; ++c) s += Bg[i * NU + c] * sm[L_DU0 + c];
      sm[L_DX + i] = s;
      xb[i] += sm[L_DX0 + i];
      lb[i] += sm[L_DL0 + i];
    }
    for (int j = threadIdx.x; j < NU; j += BLOCK) {
      ub[j]  += sm[L_DU0 + j];
      gmb[j] += sm[L_DGM + j];
    }
    __syncthreads();
    rollout(sm, Ag, Bg, Qfg, xb, ub, lb, xres, lxN, Psg, psg, Ksg, ksg, 1);
  }

  // Q0 = Lagrangian at x0 (with gmm . u0res term)
  eval_kkt(sm, Ag, Bg, Qxg, Rug, Qfg, xgg, x0b, xb, ub, lb, gmb, 0,
           lxu, lxN, xres, x0res, ls, lNp);
  float q_extra;
  {
    float part = 0.f;
    for (int j = threadIdx.x; j < NU; j += BLOCK)
      part += gmb[j] * (ub[j] - u0b[j]);
    q_extra = block_reduce(part, sm + L_RED);
  }
  const float Q0 = lagrangian(sm, ls, lNp, lb, x0res, xres, q_extra);

  // ---------------- V-iterations about x1 ----------------
  for (int it = 0; it < N_ITER; ++it) {
    eval_kkt(sm, Ag, Bg, Qxg, Rug, Qfg, xgg, x1b, xb, ub, lb, gmb, 0,
             lxu, lxN, xres, x0res, ls, lNp);
    riccati_bwd(sm, Ag, Bg, Qxg, Rug, Qfg, lxu, lxN, xres,
                Psg, psg, Ksg, ksg, 0);
    for (int i = threadIdx.x; i < NX; i += BLOCK) sm[L_DX + i] = -x0res[i];
    __syncthreads();
    rollout(sm, Ag, Bg, Qfg, xb, ub, lb, xres, lxN, Psg, psg, Ksg, ksg, 0);
  }
  eval_kkt(sm, Ag, Bg, Qxg, Rug, Qfg, xgg, x1b, xb, ub, lb, gmb, 0,
           lxu, lxN, xres, x0res, ls, lNp);
  const float V1 = lagrangian(sm, ls, lNp, lb, x0res, xres, 0.f);

  if (threadIdx.x == 0) out[b] = L0 + GAMMA_DISC * V1 - Q0;
}

// ---------------------------------------------------------------------------
extern "C" void kernel_launch(void* const* d_in, const int* in_sizes, int n_in,
                              void* d_out, int out_size, void* d_ws, size_t ws_size,
                              hipStream_t stream)
{
  const float* x0  = (const float*)d_in[0];
  const float* x1  = (const float*)d_in[1];
  const float* u0  = (const float*)d_in[2];
  const float* x   = (const float*)d_in[3];
  const float* u   = (const float*)d_in[4];
  const float* lmd = (const float*)d_in[5];
  const float* gmm = (const float*)d_in[6];
  const float* A   = (const float*)d_in[7];
  const float* Bm  = (const float*)d_in[8];
  const float* Qx  = (const float*)d_in[9];
  const float* Ru  = (const float*)d_in[10];
  const float* Qf  = (const float*)d_in[11];
  const float* xg  = (const float*)d_in[12];
  (void)in_sizes; (void)n_in; (void)out_size; (void)ws_size;

  const size_t smem = (size_t)SMEM_FLOATS * sizeof(float);   // ~220 KB (<320 KB/WGP)
  hipFuncSetAttribute(reinterpret_cast<const void*>(ocp_sqp_kernel),
                      hipFuncAttributeMaxDynamicSharedMemorySize, (int)smem);
  ocp_sqp_kernel<<<NB, BLOCK, smem, stream>>>(
      x0, x1, u0, x, u, lmd, gmm, A, Bm, Qx, Ru, Qf, xg,
      (float*)d_out, (float*)d_ws);
}